// GMP_15977278341702
// MI455X (gfx1250) — compile-verified
//
#include <hip/hip_runtime.h>

typedef float v2f __attribute__((ext_vector_type(2)));
typedef float v4f __attribute__((ext_vector_type(4)));
typedef float v8f __attribute__((ext_vector_type(8)));

#define KA 8
#define LA 8
#define KB 9            // k = 0..8
#define LB 8
#define MB 9            // m = 0..8
#define KPAD 12         // k padded to 12 per m-group == LDS row stride
#define KTOT (MB * KPAD)       // 108 (rows with k in 9..11 are zero)
#define KSTEPS 27              // 108 / 4 K-steps of the 16x16x4 f32 WMMA
#define NEDGE 8
#define WPB 8                  // waves (=16-sample tiles) per workgroup
#define CROWS 136              // window rows: 128 samples + 8 (lag or lead)
#define PSTRIDE 12

// ---------------------------------------------------------------------------
// Kernel 0: pack coefficients into the WMMA B-operand lane layout.
// B per K-step is 4x16 (KxN):
//   lanes 0-15  -> N=lane,    rows K = step*4+0 (.x), step*4+1 (.y)
//   lanes 16-31 -> N=lane-16, rows K = step*4+2 (.x), step*4+3 (.y)
// Column N<8 = Re coeff for l=N ; N>=8 = Im coeff for l=N-8.
// K flattening: Krow = m*12 + k ; k>=9 -> 0. a folded into b @ m=0.
// ---------------------------------------------------------------------------
__global__ void gmp_prep_coeffs(const float* __restrict__ a,
                                const float* __restrict__ b,
                                const float* __restrict__ c,
                                float* __restrict__ Bws) {
    int t = blockIdx.x * blockDim.x + threadIdx.x;
    if (t >= 2 * KSTEPS * 32) return;
    int term = t / (KSTEPS * 32);
    int rem  = t % (KSTEPS * 32);
    int step = rem >> 5;
    int lane = rem & 31;
    int khalf = (lane >= 16) ? 2 : 0;
    int ncol  = lane & 15;
    int l     = ncol & 7;
    int comp  = ncol >> 3;                  // 0 = Re, 1 = Im
    float vals[2];
    #pragma unroll
    for (int h = 0; h < 2; ++h) {
        int Krow = step * 4 + khalf + h;
        int m = Krow / KPAD;
        int k = Krow - m * KPAD;
        float v = 0.0f;
        if (k < KB) {
            if (term == 0) {
                v = b[(((size_t)k * LB + l) * MB + m) * 2 + comp];
                if (m == 0 && k < KA) v += a[((size_t)k * LA + l) * 2 + comp];
            } else {
                v = c[(((size_t)k * LB + l) * MB + m) * 2 + comp];
            }
        }
        vals[h] = v;
    }
    float* dst = Bws + (size_t)term * (KSTEPS * 64) + (size_t)(step * 32 + lane) * 2;
    dst[0] = vals[0];
    dst[1] = vals[1];
}

// ---------------------------------------------------------------------------
// Kernel 1: per-WG LDS-staged WMMA GEMM.
//   U[j, {Re l, Im l}] = A_b[16x108]*B_b + A_c[16x108]*B_c (f32 16x16x4 WMMA)
// Two LDS power windows:
//   sFw (forward):  row r <-> |x[clip(j0w + r)]|^k        (c-term, j+m)
//   sRv (mirrored): row r <-> |x[clip(j0w + 135 - r)]|^k  (b-term, j-m)
// With K = m*12+k == PSTRIDE, the A byte offset is linear in K, so every
// per-step A/B fetch is a ds_load_b64 at a compile-time immediate offset.
// ---------------------------------------------------------------------------
__global__ void __launch_bounds__(32 * WPB)
gmp_wmma(const float* __restrict__ xr, const float* __restrict__ xi,
         const float* __restrict__ Bws, float* __restrict__ U,
         int n, int ntiles) {
    __shared__ float sFw[CROWS * PSTRIDE];     // 6528 B
    __shared__ float sRv[CROWS * PSTRIDE];     // 6528 B
    __shared__ float sB [2 * KSTEPS * 64];     // 13824 B

    int tid  = threadIdx.x;
    int lane = tid & 31;
    int wave = tid >> 5;
    int j0w  = blockIdx.x * (WPB * 16);

    // ---- stage B panel (coalesced b128 global -> b128 LDS) ----
    #pragma unroll
    for (int i = tid; i < 2 * KSTEPS * 16; i += 32 * WPB) {   // 864 float4s
        *(v4f*)&sB[i * 4] = *(const v4f*)&Bws[i * 4];
    }
    // ---- build both power windows; union of samples: j0w-8 .. j0w+135 ----
    if (tid < 144) {
        int jj = j0w - 8 + tid;
        jj = jj < 0 ? 0 : (jj > n - 1 ? n - 1 : jj);
        float re = xr[jj], im = xi[jj];
        float r = sqrtf(re * re + im * im);
        float buf[PSTRIDE];
        float p = 1.0f;
        #pragma unroll
        for (int k = 0; k < KB; ++k) { buf[k] = p; p *= r; }
        #pragma unroll
        for (int k = KB; k < PSTRIDE; ++k) buf[k] = 0.0f;
        int rf = tid - 8;          // forward row: j = j0w + rf
        if (rf >= 0) {
            #pragma unroll
            for (int q = 0; q < 3; ++q)
                *(v4f*)&sFw[rf * PSTRIDE + q * 4] = *(v4f*)&buf[q * 4];
        }
        int rv = 135 - rf;         // mirrored row: j = j0w + 135 - rv
        if (rv <= 135) {
            #pragma unroll
            for (int q = 0; q < 3; ++q)
                *(v4f*)&sRv[rv * PSTRIDE + q * 4] = *(v4f*)&buf[q * 4];
        }
    }
    __syncthreads();

    int tile = blockIdx.x * WPB + wave;
    if (tile >= ntiles) return;                // wave-uniform, EXEC stays all-1s

    int M     = lane & 15;
    int khalf = (lane >= 16) ? 2 : 0;
    int s16   = wave * 16 + M;                 // sample within WG: jm = j0w + s16
    // loop-invariant A bases; per step the offset advances by 16 B (immediate)
    const float* baseAb = &sRv[(127 - s16) * PSTRIDE + khalf]; // row rb+m <-> jm+8-m
    const float* baseAc = &sFw[s16 * PSTRIDE + khalf];         // row fc+m <-> jm+m
    const float* baseB  = &sB [lane * 2];

    v8f acc = {0.f, 0.f, 0.f, 0.f, 0.f, 0.f, 0.f, 0.f};

    #pragma unroll
    for (int step = 0; step < KSTEPS; ++step) {            // b-term (lagging)
        v2f av = *(const v2f*)&baseAb[step * 4];
        v2f bv = *(const v2f*)&baseB [step * 64];
        acc = __builtin_amdgcn_wmma_f32_16x16x4_f32(
            false, av, false, bv, (short)0, acc, false, false);
    }
    #pragma unroll
    for (int step = 0; step < KSTEPS; ++step) {            // c-term (leading)
        v2f av = *(const v2f*)&baseAc[step * 4];
        v2f bv = *(const v2f*)&baseB [(KSTEPS + step) * 64];
        acc = __builtin_amdgcn_wmma_f32_16x16x4_f32(
            false, av, false, bv, (short)0, acc, false, false);
    }

    // D layout: VGPR i -> rows i (lanes 0-15) / i+8 (lanes 16-31), N = lane&15
    #pragma unroll
    for (int i = 0; i < 8; ++i) {
        int row = tile * 16 + i + ((lane >= 16) ? 8 : 0);
        U[(size_t)row * 16 + (lane & 15)] = acc[i];
    }
}

// ---------------------------------------------------------------------------
// Kernel 2: out[n] = sum_l x[n-l] * U[n-l, l]  (complex multiply).
// First NEDGE samples (n-l can go negative) use the direct reference formula.
// ---------------------------------------------------------------------------
__global__ void gmp_combine(const float* __restrict__ xr,
                            const float* __restrict__ xi,
                            const float* __restrict__ U,
                            const float* __restrict__ a,
                            const float* __restrict__ b,
                            const float* __restrict__ c,
                            float* __restrict__ out, int n) {
    int i = blockIdx.x * blockDim.x + threadIdx.x;
    if (i >= n) return;
    float outr = 0.f, outi = 0.f;
    if (i >= NEDGE) {
        #pragma unroll
        for (int l = 0; l < LB; ++l) {
            int j = i - l;
            float vr = xr[j], vi = xi[j];
            float ur = U[(size_t)j * 16 + l];
            float ui = U[(size_t)j * 16 + 8 + l];
            outr += vr * ur - vi * ui;
            outi += vr * ui + vi * ur;
        }
    } else {
        for (int l = 0; l < LB; ++l) {
            int jx = i - l; if (jx < 0) jx = 0;
            float vr = xr[jx], vi = xi[jx];
            {   // aligned term
                float rr = xr[jx], ri = xi[jx];
                float r = sqrtf(rr * rr + ri * ri);
                float p = 1.f;
                for (int k = 0; k < KA; ++k) {
                    float cr = a[((size_t)k * LA + l) * 2];
                    float ci = a[((size_t)k * LA + l) * 2 + 1];
                    outr += (cr * vr - ci * vi) * p;
                    outi += (cr * vi + ci * vr) * p;
                    p *= r;
                }
            }
            for (int m = 0; m < MB; ++m) {
                int jr = i - l - m; if (jr < 0) jr = 0;
                float rr = xr[jr], ri = xi[jr];
                float r = sqrtf(rr * rr + ri * ri);
                float p = 1.f;
                for (int k = 0; k < KB; ++k) {
                    float cr = b[(((size_t)k * LB + l) * MB + m) * 2];
                    float ci = b[(((size_t)k * LB + l) * MB + m) * 2 + 1];
                    outr += (cr * vr - ci * vi) * p;
                    outi += (cr * vi + ci * vr) * p;
                    p *= r;
                }
                int jc = i - l + m;
                if (jc < 0) jc = 0; if (jc > n - 1) jc = n - 1;
                rr = xr[jc]; ri = xi[jc];
                r = sqrtf(rr * rr + ri * ri);
                p = 1.f;
                for (int k = 0; k < KB; ++k) {
                    float cr = c[(((size_t)k * LB + l) * MB + m) * 2];
                    float ci = c[(((size_t)k * LB + l) * MB + m) * 2 + 1];
                    outr += (cr * vr - ci * vi) * p;
                    outi += (cr * vi + ci * vr) * p;
                    p *= r;
                }
            }
        }
    }
    out[2 * (size_t)i]     = outr;   // complex64 interleaved
    out[2 * (size_t)i + 1] = outi;
}

extern "C" void kernel_launch(void* const* d_in, const int* in_sizes, int n_in,
                              void* d_out, int out_size, void* d_ws, size_t ws_size,
                              hipStream_t stream) {
    const float* xr = (const float*)d_in[0];
    const float* xi = (const float*)d_in[1];
    const float* a  = (const float*)d_in[2];   // complex64 -> interleaved f32
    const float* b  = (const float*)d_in[3];
    const float* c  = (const float*)d_in[4];
    int n = in_sizes[0];                       // 50000
    int ntiles = (n + 15) / 16;                // 3125 (exact)
    int nwg    = (ntiles + WPB - 1) / WPB;     // 391

    float* ws  = (float*)d_ws;
    float* Bws = ws;                           // 2*27*64 = 3456 floats
    float* U   = ws + 4096;                    // ntiles*256 floats (~3 MB)

    gmp_prep_coeffs<<<(2 * KSTEPS * 32 + 255) / 256, 256, 0, stream>>>(a, b, c, Bws);
    gmp_wmma<<<nwg, 32 * WPB, 0, stream>>>(xr, xi, Bws, U, n, ntiles);
    gmp_combine<<<(n + 255) / 256, 256, 0, stream>>>(xr, xi, U, a, b, c,
                                                     (float*)d_out, n);
}